// OPMC_63685775065213
// MI455X (gfx1250) — compile-verified
//
#include <hip/hip_runtime.h>
#include <stdint.h>

// Problem shape (fixed by the reference's setup_inputs)
#define BB 4
#define CC 64
#define HH 256
#define WW 256
#define HW (HH * WW)

// Tile shape: one block = 32x8 output tile, all 5 variants, all 64 channels.
#define TX 32
#define TY 8
#define TILES_X (WW / TX)   // 8
#define TILES_Y (HH / TY)   // 32

// Flow tile in LDS: tile + 1-px halo for the 4 shifted variants, zero-padded.
#define FR (TY + 2)         // 10 rows
#define FCOL (TX + 2)       // 34 cols
#define FN (2 * FR * FCOL)  // 680 elements

// Raw LDS byte offset of a __shared__ object (AS(3) pointer value).
__device__ __forceinline__ uint32_t lds_byte_offset(const void* p) {
    return (uint32_t)(uintptr_t)(__attribute__((address_space(3))) const void*)p;
}

__global__ __launch_bounds__(256) void obmc_warp_kernel(
    const float* __restrict__ x,     // [B, C, H, W]
    const float* __restrict__ flow,  // [B, 2, H, W]
    float* __restrict__ out)         // [B, 5*C, H, W]
{
    const int tid = threadIdx.x;
    const int tx  = tid & (TX - 1);      // 0..31 (lane)
    const int ty  = tid >> 5;            // 0..7  (wave id)
    const int bx  = blockIdx.x;
    const int tile_x = bx & (TILES_X - 1);
    const int tile_y = bx >> 3;
    const int b  = blockIdx.y;

    const int x0 = tile_x * TX;
    const int y0 = tile_y * TY;
    const int px = x0 + tx;              // this thread's output column
    const int py = y0 + ty;              // this thread's output row

    // ---- Stage flow tile (+halo, zero-padded) into LDS via async DMA ----
    __shared__ float s_flow[2][FR][FCOL];
    {
        const int base_y = y0 - 1;
        const int base_x = x0 - 1;
        for (int i = tid; i < FN; i += 256) {
            const int comp = i / (FR * FCOL);
            const int rem  = i - comp * (FR * FCOL);
            const int r    = rem / FCOL;
            const int cl   = rem - r * FCOL;
            const int gy   = base_y + r;
            const int gx   = base_x + cl;
            if (gy >= 0 && gy < HH && gx >= 0 && gx < WW) {
                const float* g = flow + ((size_t)b * 2 + comp) * HW
                                      + (size_t)gy * WW + gx;
                const uint32_t loff = lds_byte_offset(&s_flow[comp][r][cl]);
                asm volatile("global_load_async_to_lds_b32 %0, %1, off"
                             :: "v"(loff), "v"(g) : "memory");
            } else {
                s_flow[comp][r][cl] = 0.0f;   // zero padding (ds_store)
            }
        }
    }
    asm volatile("s_wait_asynccnt 0" ::: "memory");
    __syncthreads();

    // ---- Per-variant taps: 4 clipped offsets + 4 masked weights each ----
    // Variant order matches the reference concat:
    //   k0: identity, k1: bias(dx=0,dy=1), k2: (1,0), k3: (0,-1), k4: (-1,0)
    const int sdx[5] = {0, 0, 1, 0, -1};
    const int sdy[5] = {0, 1, 0, -1, 0};

    int   off[5][4];
    float wgt[5][4];

    #pragma unroll
    for (int k = 0; k < 5; ++k) {
        // Shifted flow read comes straight from the zero-padded LDS tile.
        const int r  = ty + sdy[k] + 1;
        const int cl = tx + sdx[k] + 1;
        const float fdx = s_flow[0][r][cl];
        const float fdy = s_flow[1][r][cl];

        const float gx  = (float)px + fdx;
        const float gy  = (float)py + fdy;
        const float fx0 = floorf(gx);
        const float fy0 = floorf(gy);
        const float wx  = gx - fx0;
        const float wy  = gy - fy0;
        const int ix0 = (int)fx0, iy0 = (int)fy0;
        const int ix1 = ix0 + 1,  iy1 = iy0 + 1;

        const float vx0 = (ix0 >= 0 && ix0 < WW) ? 1.0f : 0.0f;
        const float vx1 = (ix1 >= 0 && ix1 < WW) ? 1.0f : 0.0f;
        const float vy0 = (iy0 >= 0 && iy0 < HH) ? 1.0f : 0.0f;
        const float vy1 = (iy1 >= 0 && iy1 < HH) ? 1.0f : 0.0f;

        wgt[k][0] = (1.0f - wx) * (1.0f - wy) * (vx0 * vy0);
        wgt[k][1] = wx          * (1.0f - wy) * (vx1 * vy0);
        wgt[k][2] = (1.0f - wx) * wy          * (vx0 * vy1);
        wgt[k][3] = wx          * wy          * (vx1 * vy1);

        const int cx0 = min(max(ix0, 0), WW - 1);
        const int cx1 = min(max(ix1, 0), WW - 1);
        const int cy0 = min(max(iy0, 0), HH - 1);
        const int cy1 = min(max(iy1, 0), HH - 1);

        off[k][0] = cy0 * WW + cx0;
        off[k][1] = cy0 * WW + cx1;
        off[k][2] = cy1 * WW + cx0;
        off[k][3] = cy1 * WW + cx1;
    }

    const float* xb    = x   + (size_t)b * CC * HW;
    float*       obase = out + (size_t)b * (5 * CC) * HW
                             + (size_t)py * WW + px;

    // ---- Channel-major sweep: one footprint fetch serves all 5 variants ----
    #pragma unroll 2
    for (int c = 0; c < CC; ++c) {
        const float* xc = xb + (size_t)c * HW;
        #pragma unroll
        for (int k = 0; k < 5; ++k) {
            float v = wgt[k][0] * xc[off[k][0]];
            v = fmaf(wgt[k][1], xc[off[k][1]], v);
            v = fmaf(wgt[k][2], xc[off[k][2]], v);
            v = fmaf(wgt[k][3], xc[off[k][3]], v);
            __builtin_nontemporal_store(v, obase + (size_t)(k * CC + c) * HW);
        }
    }
}

extern "C" void kernel_launch(void* const* d_in, const int* in_sizes, int n_in,
                              void* d_out, int out_size, void* d_ws, size_t ws_size,
                              hipStream_t stream) {
    (void)in_sizes; (void)n_in; (void)out_size; (void)d_ws; (void)ws_size;
    const float* x    = (const float*)d_in[0];   // [4,64,256,256]
    const float* flow = (const float*)d_in[1];   // [4,2,256,256]
    float* out        = (float*)d_out;           // [4,320,256,256]

    dim3 grid(TILES_X * TILES_Y, BB);  // 256 tiles x 4 batches = 1024 blocks
    dim3 block(256);                   // 8 wave32 per block
    obmc_warp_kernel<<<grid, block, 0, stream>>>(x, flow, out);
}